// BC_evidences_X_56358560858217
// MI455X (gfx1250) — compile-verified
//
#include <hip/hip_runtime.h>
#include <cmath>

// Problem constants (from reference)
#define N_NODES 100000
#define T_STEPS 512
#define E_EDGES 1024
#define MU 0.05f
#define RHO 32.0f
// Per-node event capacity: expected events/node ~5 (Poisson), CAP=48 gives huge margin.
#define CAP 48

typedef __attribute__((ext_vector_type(4))) unsigned int v4u;
typedef __attribute__((ext_vector_type(8))) int v8i;
typedef __attribute__((ext_vector_type(4))) int v4i;

// Fast sigmoid: v_exp_f32 + v_rcp_f32 (1-ulp rcp instead of the ~15-op IEEE
// division expansion the default codegen produced).
__device__ __forceinline__ float sigmoidf(float x) {
    return __builtin_amdgcn_rcpf(1.0f + __expf(-x));
}

// ---------------------------------------------------------------------------
// TDM: async DMA of one step's edge tile (E*3 = 3072 dwords = 12KB) into LDS.
// Descriptor per CDNA5 ISA §8 (D# group0/group1), 1D tile: tensor_dim0 =
// tile_dim0 = 3072 dwords, data_size = 4B, no padding / iterate / multicast.
// This toolchain (clang-23 / therock-10.0) uses the 6-arg builtin form.
// ---------------------------------------------------------------------------
__device__ __forceinline__ void tdm_load_edges(unsigned lds_byte_off, const void* gptr) {
    unsigned long long ga = (unsigned long long)(uintptr_t)gptr;
    const unsigned DIM0 = (unsigned)(E_EDGES * 3); // 3072 dwords
    v4u g0 = {
        1u,                                   // count=1, user mode, no gather
        lds_byte_off,                         // lds_addr (bytes)
        (unsigned)(ga & 0xFFFFFFFFull),       // global_addr[31:0]
        (unsigned)((ga >> 32) & 0x1FFFFFFull) // global_addr[56:32]
            | (2u << 30)                      // type = 2 ("image")
    };
    v8i g1 = {
        (int)(2u << 16),                      // wg_mask=0, data_size=2 (4B)
        (int)((DIM0 & 0xFFFFu) << 16),        // tensor_dim0[15:0]
        (int)((DIM0 >> 16) | (1u << 16)),     // tensor_dim0[31:16] | tensor_dim1_lo=1
        (int)(0u | (DIM0 << 16)),             // tensor_dim1_hi=0 | tile_dim0=3072
        0,                                    // tile_dim1=0 (unused), tile_dim2=0
        (int)DIM0,                            // tensor_dim0_stride[31:0]
        0,                                    // stride0_hi | stride1_lo
        0                                     // stride1_hi
    };
    v4i gz4 = {0, 0, 0, 0};
    v8i gz8 = {0, 0, 0, 0, 0, 0, 0, 0};
    // amdgpu-toolchain 6-arg form: (g0, g1, g2, g3, g4, cpol)
    __builtin_amdgcn_tensor_load_to_lds(g0, g1, gz4, gz4, gz8, 0);
}

// Read one edge record (u,v,s) from the DMA-written LDS tile via explicit
// ds_load_b32. The tile is written only by the TDM (invisible to the
// compiler), so the loads + their DScnt wait live inside one asm block.
__device__ __forceinline__ void lds_read_edge(unsigned byte_off, int& u, int& v, int& s) {
    asm volatile(
        "ds_load_b32 %0, %3 offset:0\n\t"
        "ds_load_b32 %1, %3 offset:4\n\t"
        "ds_load_b32 %2, %3 offset:8\n\t"
        "s_wait_dscnt 0x0"
        : "=&v"(u), "=&v"(v), "=&v"(s)
        : "v"(byte_off)
        : "memory");
}

// ---------------------------------------------------------------------------
// Kernel 1: X0 = sigmoid(logit_X0); init scan state, event counters, out row 0
// ---------------------------------------------------------------------------
__global__ void k_init(const float* __restrict__ logit, float* __restrict__ Xcur,
                       int* __restrict__ cnt, float* __restrict__ outX) {
    int i = blockIdx.x * blockDim.x + threadIdx.x;
    if (i < N_NODES) {
        float x0 = sigmoidf(logit[i]);
        Xcur[i]  = x0;
        outX[i]  = x0;   // X[0][i]
        cnt[i]   = 0;
    }
}

// ---------------------------------------------------------------------------
// Kernel 2: the sequential scan. One workgroup, 1024 threads (1 thread/edge).
// State Xcur lives in global memory (400KB -> L2 resident). Edge tiles are
// streamed into LDS with the Tensor Data Mover, double-buffered; reads come
// back via explicit ds_load_b32 (see lds_read_edge).
// Emits kappa and sparse per-node (step, delta) events.
// ---------------------------------------------------------------------------
__global__ void __launch_bounds__(1024) k_scan(
    const int* __restrict__ edges, const float* __restrict__ theta,
    float* __restrict__ Xcur, int* __restrict__ cnt,
    int* __restrict__ ev_step, float* __restrict__ ev_delta,
    float* __restrict__ kappa)
{
    // 24KB tile double-buffer at LDS offset 0 (only shared object in kernel).
    __shared__ __align__(16) int s_edges[2][E_EDGES * 3];
    (void)s_edges; // written by TDM, read by inline-asm ds_load
    const int tid = threadIdx.x;
    const float eps = sigmoidf(theta[0]);

    if (tid == 0) {
        tdm_load_edges(0u, edges);            // preload step 0 into buffer 0
    }

    for (int t = 0; t < T_STEPS - 1; ++t) {
        const int b = t & 1;
        // Wave 0 waits for its TDM; other waves have TENSORcnt==0 (no-op).
        __builtin_amdgcn_s_wait_tensorcnt((short)0);
        __syncthreads();                      // tile visible to whole workgroup

        // Kick next tile while this step computes (double buffer).
        if (tid == 0 && (t + 1) < (T_STEPS - 1)) {
            tdm_load_edges((unsigned)((b ^ 1) * (E_EDGES * 3 * 4)),
                           edges + (size_t)(t + 1) * E_EDGES * 3);
        }

        int u, v, s;
        lds_read_edge((unsigned)(b * (E_EDGES * 3 * 4) + tid * 12), u, v, s);

        // Gather pre-update state (agent scope: must see prior steps' atomics).
        float xu = __hip_atomic_load(&Xcur[u], __ATOMIC_RELAXED, __HIP_MEMORY_SCOPE_AGENT);
        float xv = __hip_atomic_load(&Xcur[v], __ATOMIC_RELAXED, __HIP_MEMORY_SCOPE_AGENT);
        float diff = xu - xv;

        // kappa uses pre-update diff, independent of s.
        kappa[(size_t)t * E_EDGES + tid] = sigmoidf(RHO * (eps - fabsf(diff)));

        // All gathers of this step must land before any scatter of this step.
        asm volatile("s_wait_loadcnt 0x0" ::: "memory");
        __syncthreads();

        if (s != 0) {
            float du = -MU * diff;            // X[u] += -MU*d, X[v] += +MU*d
            unsafeAtomicAdd(&Xcur[u], du);
            unsafeAtomicAdd(&Xcur[v], -du);
            int su = atomicAdd(&cnt[u], 1);
            if (su < CAP) {
                ev_step [(size_t)u * CAP + su] = t;
                ev_delta[(size_t)u * CAP + su] = du;
            }
            int sv = atomicAdd(&cnt[v], 1);
            if (sv < CAP) {
                ev_step [(size_t)v * CAP + sv] = t;
                ev_delta[(size_t)v * CAP + sv] = -du;
            }
        }
        // All scatters at coherence point before next step's gathers.
        asm volatile("s_wait_storecnt 0x0" ::: "memory");
        __syncthreads();
    }
}

// ---------------------------------------------------------------------------
// Kernel 3: materialize X (T x N). Thread n owns column n, replays its sparse
// event list while sweeping t. Stores are coalesced across lanes (consecutive
// n for the same t) and nontemporal (write-once stream, 205MB).
// ---------------------------------------------------------------------------
__global__ void k_materialize(const int* __restrict__ cnt,
                              const int* __restrict__ ev_step,
                              const float* __restrict__ ev_delta,
                              float* __restrict__ outX)
{
    int n = blockIdx.x * blockDim.x + threadIdx.x;
    if (n >= N_NODES) return;
    float cur = outX[n];                       // row 0 = X0
    int c = cnt[n]; if (c > CAP) c = CAP;
    int p = 0;
    const int*   es = ev_step  + (size_t)n * CAP;
    const float* ed = ev_delta + (size_t)n * CAP;
    for (int t = 1; t < T_STEPS; ++t) {
        while (p < c && es[p] < t) { cur += ed[p]; ++p; }
        __builtin_nontemporal_store(cur, &outX[(size_t)t * N_NODES + n]);
    }
}

// ---------------------------------------------------------------------------
extern "C" void kernel_launch(void* const* d_in, const int* in_sizes, int n_in,
                              void* d_out, int out_size, void* d_ws, size_t ws_size,
                              hipStream_t stream)
{
    const float* logit = (const float*)d_in[0];   // (N,)
    const float* theta = (const float*)d_in[1];   // (1,)
    const int*   edges = (const int*)d_in[2];     // (T-1, E, 3)

    float* outX  = (float*)d_out;                       // (T, N)
    float* kappa = outX + (size_t)T_STEPS * N_NODES;    // ((T-1)*E,)

    // Workspace layout (all 16B-aligned: N*4 = 400000 is a multiple of 16)
    float* Xcur    = (float*)d_ws;
    int*   cnt     = (int*)(Xcur + N_NODES);
    int*   ev_step = cnt + N_NODES;
    float* ev_delta= (float*)(ev_step + (size_t)N_NODES * CAP);
    // Total: 2*N*4 + 2*N*CAP*4 ~= 39.2 MB

    k_init<<<(N_NODES + 255) / 256, 256, 0, stream>>>(logit, Xcur, cnt, outX);
    k_scan<<<1, 1024, 0, stream>>>(edges, theta, Xcur, cnt, ev_step, ev_delta, kappa);
    k_materialize<<<(N_NODES + 255) / 256, 256, 0, stream>>>(cnt, ev_step, ev_delta, outX);
}